// ForwardEuler_79182017069697
// MI455X (gfx1250) — compile-verified
//
#include <hip/hip_runtime.h>
#include <math.h>

// Problem constants (match the reference)
#define BATCH   256
#define STATE   512
#define HIDDEN  1024
#define COND    64
#define NSTEPS  256
#define TSCALE  0.1f

typedef __attribute__((ext_vector_type(2))) float v2f;
typedef __attribute__((ext_vector_type(4))) float v4f;
typedef __attribute__((ext_vector_type(8))) float v8f;

static __device__ __forceinline__ v8f wmma_f32(v2f a, v2f b, v8f c) {
    return __builtin_amdgcn_wmma_f32_16x16x4_f32(
        /*neg_a=*/false, a, /*neg_b=*/false, b,
        /*c_mod=*/(short)0, c, /*reuse_a=*/false, /*reuse_b=*/false);
}

// ---------------------------------------------------------------------------
// Copy x0 into output slice 0 (output doubles as the state ping-pong buffer).
// ---------------------------------------------------------------------------
__global__ __launch_bounds__(256) void copy_x0_kernel(const float* __restrict__ x0,
                                                      float* __restrict__ out) {
    int i = blockIdx.x * 256 + threadIdx.x;
    out[i] = x0[i];
}

// ---------------------------------------------------------------------------
// Precompute (once): Z1[b,j] = b1[j] + sum_k z[b,k] * W1[512+k, j]
// ---------------------------------------------------------------------------
__global__ __launch_bounds__(256) void z1_precompute_kernel(const float* __restrict__ z,
                                                            const float* __restrict__ W1,
                                                            const float* __restrict__ b1,
                                                            float* __restrict__ Z1) {
    int idx = blockIdx.x * 256 + threadIdx.x;     // 0 .. BATCH*HIDDEN-1
    int b = idx >> 10;                            // HIDDEN == 1024
    int j = idx & (HIDDEN - 1);
    float acc = b1[j];
    const float* zrow = z + b * COND;
    const float* wz   = W1 + (size_t)STATE * HIDDEN + j;  // rows 512..575
    #pragma unroll 4
    for (int k = 0; k < COND; ++k)
        acc += zrow[k] * wz[(size_t)k * HIDDEN];
    Z1[idx] = acc;
}

// ---------------------------------------------------------------------------
// One-time weight repack into B-fragment-major order.
// For each n-tile nt and 8-wide k-block kb2, lane L's 4 floats (the B frags
// for two consecutive 16x16x4 WMMAs) are contiguous:
//   Wp[((nt*(K/8)+kb2)*32 + L)*4 + {0,1,2,3}]
//     = W[k0+ak+0][n], W[k0+ak+1][n], W[k0+4+ak+0][n], W[k0+4+ak+1][n]
// with k0 = kb2*8, ak = (L>>4)*2, n = nt*16 + (L&15).
// Inner loops then read one coalesced global_load_b128 per 2 WMMAs.
// ---------------------------------------------------------------------------
__global__ __launch_bounds__(256) void pack_w_kernel(const float* __restrict__ W,
                                                     float* __restrict__ Wp,
                                                     int K, int N) {
    int idx  = blockIdx.x * 256 + threadIdx.x;    // total (N/16)*(K/8)*32 threads
    int lane = idx & 31;
    int kb2  = (idx >> 5) % (K >> 3);
    int nt   = (idx >> 5) / (K >> 3);
    int k0   = (kb2 << 3) + ((lane >> 4) << 1);
    int n    = (nt << 4) + (lane & 15);
    size_t o = (size_t)idx * 4;
    Wp[o + 0] = W[(size_t)(k0 + 0) * N + n];
    Wp[o + 1] = W[(size_t)(k0 + 1) * N + n];
    Wp[o + 2] = W[(size_t)(k0 + 4) * N + n];
    Wp[o + 3] = W[(size_t)(k0 + 5) * N + n];
}

// ---------------------------------------------------------------------------
// Packed step kernel 1:  H = tanh( X @ W1x + Z1 + t_s * W1t )
// Grid: 16 M-tiles x 8 N-groups = 128 blocks, 8 waves/block, 1 tile/wave.
// ---------------------------------------------------------------------------
__global__ __launch_bounds__(256) void step1_packed_kernel(const float* __restrict__ X,
                                                           const float* __restrict__ W1pack,
                                                           const float* __restrict__ W1,
                                                           const float* __restrict__ Z1,
                                                           const float* __restrict__ t,
                                                           int s,
                                                           float* __restrict__ Hbuf) {
    __shared__ float xs[16 * STATE];              // 32 KB

    const int mt   = blockIdx.x >> 3;
    const int ng   = blockIdx.x & 7;
    const int m0   = mt * 16;
    const int wave = threadIdx.x >> 5;
    const int lane = threadIdx.x & 31;
    const int n0   = (ng * 8 + wave) * 16;

    for (int i = threadIdx.x; i < 16 * STATE; i += 256)
        xs[i] = X[(size_t)(m0 + (i >> 9)) * STATE + (i & (STATE - 1))];
    __syncthreads();

    const int ar = lane & 15;
    const int ak = (lane >> 4) * 2;
    const float* wp = W1pack + ((size_t)(n0 >> 4) * (STATE >> 3)) * 128 + lane * 4;

    v8f acc = {0.f, 0.f, 0.f, 0.f, 0.f, 0.f, 0.f, 0.f};
    for (int kb = 0; kb < (STATE >> 3); ++kb) {   // 64 iterations, 2 WMMAs each
        v4f w  = *(const v4f*)(wp + (size_t)kb * 128);
        v2f a0 = *(const v2f*)&xs[ar * STATE + kb * 8 + ak];
        v2f a1 = *(const v2f*)&xs[ar * STATE + kb * 8 + 4 + ak];
        v2f b0 = {w.x, w.y};
        v2f b1 = {w.z, w.w};
        acc = wmma_f32(a0, b0, acc);
        acc = wmma_f32(a1, b1, acc);
    }

    const float ts   = t[s];
    const float* W1t = W1 + (size_t)(STATE + COND) * HIDDEN;  // row 576

    const int col   = n0 + (lane & 15);
    const int rbase = m0 + ((lane >> 4) << 3);
    #pragma unroll
    for (int r = 0; r < 8; ++r) {
        const int row = rbase + r;
        float v = acc[r] + Z1[(size_t)row * HIDDEN + col] + ts * W1t[col];
        Hbuf[(size_t)row * HIDDEN + col] = tanhf(v);
    }
}

// ---------------------------------------------------------------------------
// Packed step kernel 2:  Xnext = relu( X + dt * (H @ W2 + b2) )
// Grid: 16 M-tiles x 4 N-groups = 64 blocks, 8 waves/block, 1 tile/wave.
// ---------------------------------------------------------------------------
__global__ __launch_bounds__(256) void step2_packed_kernel(const float* __restrict__ Hbuf,
                                                           const float* __restrict__ W2pack,
                                                           const float* __restrict__ b2,
                                                           const float* __restrict__ X,
                                                           const float* __restrict__ t,
                                                           float* __restrict__ Xnext) {
    __shared__ float hs[16 * HIDDEN];             // 64 KB

    const int mt   = blockIdx.x >> 2;
    const int ng   = blockIdx.x & 3;
    const int m0   = mt * 16;
    const int wave = threadIdx.x >> 5;
    const int lane = threadIdx.x & 31;
    const int n0   = (ng * 8 + wave) * 16;

    for (int i = threadIdx.x; i < 16 * HIDDEN; i += 256)
        hs[i] = Hbuf[(size_t)(m0 + (i >> 10)) * HIDDEN + (i & (HIDDEN - 1))];
    __syncthreads();

    const int ar = lane & 15;
    const int ak = (lane >> 4) * 2;
    const float* wp = W2pack + ((size_t)(n0 >> 4) * (HIDDEN >> 3)) * 128 + lane * 4;

    v8f acc = {0.f, 0.f, 0.f, 0.f, 0.f, 0.f, 0.f, 0.f};
    for (int kb = 0; kb < (HIDDEN >> 3); ++kb) {  // 128 iterations, 2 WMMAs each
        v4f w  = *(const v4f*)(wp + (size_t)kb * 128);
        v2f a0 = *(const v2f*)&hs[ar * HIDDEN + kb * 8 + ak];
        v2f a1 = *(const v2f*)&hs[ar * HIDDEN + kb * 8 + 4 + ak];
        v2f b0 = {w.x, w.y};
        v2f b1 = {w.z, w.w};
        acc = wmma_f32(a0, b0, acc);
        acc = wmma_f32(a1, b1, acc);
    }

    const float dt = (t[1] - t[0]) * TSCALE;

    const int col   = n0 + (lane & 15);
    const int rbase = m0 + ((lane >> 4) << 3);
    #pragma unroll
    for (int r = 0; r < 8; ++r) {
        const int row = rbase + r;
        float f = acc[r] + b2[col];
        float v = X[(size_t)row * STATE + col] + dt * f;
        Xnext[(size_t)row * STATE + col] = fmaxf(v, 0.f);
    }
}

// ---------------------------------------------------------------------------
// Fallback (no weight repack, reads W directly): used only if d_ws is too
// small to hold the packed weights.
// ---------------------------------------------------------------------------
__global__ __launch_bounds__(256) void step1_direct_kernel(const float* __restrict__ X,
                                                           const float* __restrict__ W1,
                                                           const float* __restrict__ Z1,
                                                           const float* __restrict__ t,
                                                           int s,
                                                           float* __restrict__ Hbuf) {
    __shared__ float xs[16 * STATE];
    const int mt = blockIdx.x >> 3, ng = blockIdx.x & 7;
    const int m0 = mt * 16;
    const int wave = threadIdx.x >> 5, lane = threadIdx.x & 31;
    const int n0 = (ng * 8 + wave) * 16;
    for (int i = threadIdx.x; i < 16 * STATE; i += 256)
        xs[i] = X[(size_t)(m0 + (i >> 9)) * STATE + (i & (STATE - 1))];
    __syncthreads();
    const int ar = lane & 15, ak = (lane >> 4) * 2;
    v8f acc = {0.f, 0.f, 0.f, 0.f, 0.f, 0.f, 0.f, 0.f};
    for (int k = 0; k < STATE; k += 4) {
        v2f a = *(const v2f*)&xs[ar * STATE + k + ak];
        v2f b;
        b.x = W1[(size_t)(k + ak) * HIDDEN + n0 + ar];
        b.y = W1[(size_t)(k + ak + 1) * HIDDEN + n0 + ar];
        acc = wmma_f32(a, b, acc);
    }
    const float ts = t[s];
    const float* W1t = W1 + (size_t)(STATE + COND) * HIDDEN;
    const int col = n0 + (lane & 15);
    const int rbase = m0 + ((lane >> 4) << 3);
    #pragma unroll
    for (int r = 0; r < 8; ++r) {
        const int row = rbase + r;
        float v = acc[r] + Z1[(size_t)row * HIDDEN + col] + ts * W1t[col];
        Hbuf[(size_t)row * HIDDEN + col] = tanhf(v);
    }
}

__global__ __launch_bounds__(256) void step2_direct_kernel(const float* __restrict__ Hbuf,
                                                           const float* __restrict__ W2,
                                                           const float* __restrict__ b2,
                                                           const float* __restrict__ X,
                                                           const float* __restrict__ t,
                                                           float* __restrict__ Xnext) {
    __shared__ float hs[16 * HIDDEN];
    const int mt = blockIdx.x >> 2, ng = blockIdx.x & 3;
    const int m0 = mt * 16;
    const int wave = threadIdx.x >> 5, lane = threadIdx.x & 31;
    const int n0 = (ng * 8 + wave) * 16;
    for (int i = threadIdx.x; i < 16 * HIDDEN; i += 256)
        hs[i] = Hbuf[(size_t)(m0 + (i >> 10)) * HIDDEN + (i & (HIDDEN - 1))];
    __syncthreads();
    const int ar = lane & 15, ak = (lane >> 4) * 2;
    v8f acc = {0.f, 0.f, 0.f, 0.f, 0.f, 0.f, 0.f, 0.f};
    for (int k = 0; k < HIDDEN; k += 4) {
        v2f a = *(const v2f*)&hs[ar * HIDDEN + k + ak];
        v2f b;
        b.x = W2[(size_t)(k + ak) * STATE + n0 + ar];
        b.y = W2[(size_t)(k + ak + 1) * STATE + n0 + ar];
        acc = wmma_f32(a, b, acc);
    }
    const float dt = (t[1] - t[0]) * TSCALE;
    const int col = n0 + (lane & 15);
    const int rbase = m0 + ((lane >> 4) << 3);
    #pragma unroll
    for (int r = 0; r < 8; ++r) {
        const int row = rbase + r;
        float f = acc[r] + b2[col];
        float v = X[(size_t)row * STATE + col] + dt * f;
        Xnext[(size_t)row * STATE + col] = fmaxf(v, 0.f);
    }
}

// ---------------------------------------------------------------------------
// Launch. d_out slice s == state at step s, so d_out is the state chain.
// ---------------------------------------------------------------------------
extern "C" void kernel_launch(void* const* d_in, const int* in_sizes, int n_in,
                              void* d_out, int out_size, void* d_ws, size_t ws_size,
                              hipStream_t stream) {
    const float* x0 = (const float*)d_in[0];
    const float* z  = (const float*)d_in[1];
    const float* t  = (const float*)d_in[2];
    const float* W1 = (const float*)d_in[3];
    const float* b1 = (const float*)d_in[4];
    const float* W2 = (const float*)d_in[5];
    const float* b2 = (const float*)d_in[6];
    float* out = (float*)d_out;

    const size_t nZ1 = (size_t)BATCH * HIDDEN;        // 262144 floats
    const size_t nH  = (size_t)BATCH * HIDDEN;        // 262144 floats
    const size_t nW1 = (size_t)STATE * HIDDEN;        // 524288 floats
    const size_t nW2 = (size_t)HIDDEN * STATE;        // 524288 floats

    float* Z1     = (float*)d_ws;
    float* Hbuf   = Z1 + nZ1;
    float* W1pack = Hbuf + nH;
    float* W2pack = W1pack + nW1;

    const bool packed = ws_size >= (nZ1 + nH + nW1 + nW2) * sizeof(float);

    copy_x0_kernel<<<(BATCH * STATE) / 256, 256, 0, stream>>>(x0, out);
    z1_precompute_kernel<<<(BATCH * HIDDEN) / 256, 256, 0, stream>>>(z, W1, b1, Z1);

    if (packed) {
        // (N/16)*(K/8)*32 threads each
        pack_w_kernel<<<(int)((nW1 / 4) / 256), 256, 0, stream>>>(W1, W1pack, STATE, HIDDEN);
        pack_w_kernel<<<(int)((nW2 / 4) / 256), 256, 0, stream>>>(W2, W2pack, HIDDEN, STATE);
        for (int s = 0; s < NSTEPS - 1; ++s) {
            const float* xcur  = out + (size_t)s * BATCH * STATE;
            float*       xnext = out + (size_t)(s + 1) * BATCH * STATE;
            step1_packed_kernel<<<128, 256, 0, stream>>>(xcur, W1pack, W1, Z1, t, s, Hbuf);
            step2_packed_kernel<<<64, 256, 0, stream>>>(Hbuf, W2pack, b2, xcur, t, xnext);
        }
    } else {
        for (int s = 0; s < NSTEPS - 1; ++s) {
            const float* xcur  = out + (size_t)s * BATCH * STATE;
            float*       xnext = out + (size_t)(s + 1) * BATCH * STATE;
            step1_direct_kernel<<<128, 256, 0, stream>>>(xcur, W1, Z1, t, s, Hbuf);
            step2_direct_kernel<<<64, 256, 0, stream>>>(Hbuf, W2, b2, xcur, t, xnext);
        }
    }
}